// S1_to_S2_67568425501190
// MI455X (gfx1250) — compile-verified
//
#include <hip/hip_runtime.h>
#include <hip/hip_bf16.h>
#include <cstdint>

typedef __attribute__((ext_vector_type(2))) float v2f;
typedef __attribute__((ext_vector_type(8))) float v8f;

#define KTOT 65536   // D*T
#define DIMC 128
#define KSPLIT 8
#define KSEG (KTOT / KSPLIT)   // 8192 = 128 chunks of 64

// ---------------------------------------------------------------------------
// Kernel 1: partial GEMM  part[seg, m, j] = sum_{k in seg} A[m,k] * We[pk, j]
// A[m,kk] = x[n,d,t,v] with m=n*NV+v, kk=d*256+t (K reordered for coalescing;
// B row permuted to match: refk = t*256+d). Block = 16 M-rows x one K-segment,
// 8 waves each owning a 16-wide N tile; A chunk (16x64) staged in LDS; 16
// v_wmma_f32_16x16x4_f32 per chunk. K-split x8 -> 480 blocks to fill the chip
// (the stage streams ~318 MB and must engage all WGPs' load paths).
// ---------------------------------------------------------------------------
template<int NV>
__global__ __launch_bounds__(256)
void xa_gemm_kernel(const float* __restrict__ x,
                    const float* __restrict__ We,
                    float* __restrict__ part)     // (KSPLIT, 96*NV, 128)
{
    __shared__ float As[16][65];          // pad 65: conflict-free reads
    const int m0   = blockIdx.x * 16;
    const int seg  = blockIdx.y;
    const int lane = threadIdx.x & 31;
    const int wave = threadIdx.x >> 5;
    const int j0   = wave * 16;
    const int mrow = lane & 15;
    const int r0k  = (lane >> 4) * 2;     // A frag: lanes 16-31 hold K=2,3

    // precompute cooperative A-stage offsets (4 elements per thread)
    size_t aoff[4]; int ar[4], ac[4];
#pragma unroll
    for (int ee = 0; ee < 4; ++ee) {
        int e = threadIdx.x + ee * 256;
        int r = e >> 6, c = e & 63;
        int m = m0 + r;
        int n = m / NV;
        int v = m - n * NV;
        ar[ee] = r; ac[ee] = c;
        aoff[ee] = (size_t)n * (size_t)KTOT * NV + (size_t)c * NV + v;
    }
    const size_t boff = (size_t)r0k * 32768 + (size_t)j0 + (lane & 15);

    v8f acc = {0.f,0.f,0.f,0.f,0.f,0.f,0.f,0.f};

    const int kbeg = seg * KSEG, kend = kbeg + KSEG;
    for (int kk0 = kbeg; kk0 < kend; kk0 += 64) {
        const int d  = kk0 >> 8;
        const int t0 = kk0 & 255;
#pragma unroll
        for (int ee = 0; ee < 4; ++ee)
            As[ar[ee]][ac[ee]] = x[aoff[ee] + (size_t)kk0 * NV];
        if (kk0 + 64 < kend)               // prefetch next A chunk
            __builtin_prefetch(&x[aoff[0] + (size_t)(kk0 + 64) * NV], 0, 1);
        __syncthreads();

        const float* bbase = We + (size_t)t0 * 32768 + (size_t)d * 128 + boff;
#pragma unroll
        for (int s = 0; s < 16; ++s) {
            v2f a, b;
            a.x = As[mrow][4*s + r0k];
            a.y = As[mrow][4*s + r0k + 1];
            b.x = bbase[(size_t)s * 131072];
            b.y = bbase[(size_t)s * 131072 + 32768];
            acc = __builtin_amdgcn_wmma_f32_16x16x4_f32(
                false, a, false, b, (short)0, acc, false, false);
        }
        __syncthreads();
    }

    // D layout: lane L, vgpr r -> row = r + (L<16?0:8), col = L&15
    const int mhi = (lane >> 4) * 8;
    float* pb = part + (size_t)seg * (96 * NV) * DIMC;
#pragma unroll
    for (int r = 0; r < 8; ++r)
        pb[(size_t)(m0 + mhi + r) * DIMC + j0 + (lane & 15)] = acc[r];
}

// ---------------------------------------------------------------------------
// Kernel 1b: deterministic segment reduction + relu.  total = 96*NV*128.
// ---------------------------------------------------------------------------
__global__ void xa_reduce_kernel(const float* __restrict__ part,
                                 float* __restrict__ xa, int total)
{
    int idx = blockIdx.x * blockDim.x + threadIdx.x;
    if (idx < total) {
        float s = 0.f;
#pragma unroll
        for (int g = 0; g < KSPLIT; ++g)
            s += part[(size_t)g * total + idx];
        xa[idx] = fmaxf(s, 0.f);
    }
}

// ---------------------------------------------------------------------------
// Kernel 2: pair attentions. att12/att21[p,k,v] = softmax_v(relu(w @ cat)+.5A)
// cat[p,v,:] = [xa[2p,v,:], xa[2p+1,v,:]]
// ---------------------------------------------------------------------------
template<int NV>
__global__ void pair_att_kernel(const float* __restrict__ xa,
                                const float* __restrict__ w12,
                                const float* __restrict__ w21,
                                const float* __restrict__ A,
                                float* __restrict__ att12,
                                float* __restrict__ att21)
{
    const int p = blockIdx.x;
    __shared__ float lg[2][NV][NV];
    const int tid = threadIdx.x;
    if (tid < 2 * NV * NV) {
        int which = tid / (NV * NV);
        int r = tid % (NV * NV);
        int k = r / NV, v = r % NV;
        const float* w  = which ? w21 : w12;                   // (NV, 256)
        const float* xe = xa + ((size_t)(2*p)   * NV + v) * DIMC;
        const float* xo = xa + ((size_t)(2*p+1) * NV + v) * DIMC;
        float s = 0.f;
        for (int c = 0; c < DIMC; ++c) s += w[k*256 + c]        * xe[c];
        for (int c = 0; c < DIMC; ++c) s += w[k*256 + DIMC + c] * xo[c];
        lg[which][k][v] = fmaxf(s, 0.f) + 0.5f * A[k*NV + v];
    }
    __syncthreads();
    if (tid < 2 * NV * NV) {
        int which = tid / (NV * NV);
        int r = tid % (NV * NV);
        int k = r / NV, v = r % NV;
        float mx = -1e30f;
        for (int u = 0; u < NV; ++u) mx = fmaxf(mx, lg[which][k][u]);
        float den = 0.f;
        for (int u = 0; u < NV; ++u) den += __expf(lg[which][k][u] - mx);
        float val = __expf(lg[which][k][v] - mx) / den;
        (which ? att21 : att12)[((size_t)p*NV + k)*NV + v] = val;
    }
}

// ---------------------------------------------------------------------------
// Kernel 3: joint attentions (no relu!):
// att [n,w,v] = softmax_v( xa2[n,w,:] @ weight  @ xa1[n ,v,:] + .5*Aj )
// att2[n,w,v] = softmax_v( xa2[n,w,:] @ weight2 @ xa1[pn,v,:] + .5*Aj )
// pn = n<48 ? 2n+1 : 2(n-48)     (the t2t1-swapped xa1)
// ---------------------------------------------------------------------------
__global__ void joint_att_kernel(const float* __restrict__ xa1,
                                 const float* __restrict__ xa2,
                                 const float* __restrict__ weight,
                                 const float* __restrict__ weight2,
                                 const float* __restrict__ Aj,
                                 float* __restrict__ att,
                                 float* __restrict__ att2)
{
    const int n  = blockIdx.x;
    const int pn = (n < 48) ? (2*n + 1) : (2*(n - 48));
    __shared__ float z[3][DIMC], z2[3][DIMC];
    __shared__ float lg[2][3][7];
    const int d = threadIdx.x;                 // 128 threads
    const float* x2 = xa2 + (size_t)n * 3 * DIMC;
    {
        float s[3] = {0,0,0}, s2[3] = {0,0,0};
        for (int c = 0; c < DIMC; ++c) {
            float wv  = weight [c*DIMC + d];
            float wv2 = weight2[c*DIMC + d];
#pragma unroll
            for (int w = 0; w < 3; ++w) {
                float a = x2[w*DIMC + c];
                s[w] += a * wv;  s2[w] += a * wv2;
            }
        }
#pragma unroll
        for (int w = 0; w < 3; ++w) { z[w][d] = s[w]; z2[w][d] = s2[w]; }
    }
    __syncthreads();
    if (threadIdx.x < 42) {
        int which = threadIdx.x / 21;
        int r = threadIdx.x % 21;
        int w = r / 7, v = r % 7;
        const float* x1 = xa1 + ((size_t)(which ? pn : n) * 7 + v) * DIMC;
        const float* zz = which ? &z2[w][0] : &z[w][0];
        float s = 0.f;
        for (int c = 0; c < DIMC; ++c) s += zz[c] * x1[c];
        lg[which][w][v] = s + 0.5f * Aj[w*7 + v];
    }
    __syncthreads();
    if (threadIdx.x < 42) {
        int which = threadIdx.x / 21;
        int r = threadIdx.x % 21;
        int w = r / 7, v = r % 7;
        float mx = -1e30f;
        for (int u = 0; u < 7; ++u) mx = fmaxf(mx, lg[which][w][u]);
        float den = 0.f;
        for (int u = 0; u < 7; ++u) den += __expf(lg[which][w][u] - mx);
        float val = __expf(lg[which][w][v] - mx) / den;
        (which ? att2 : att)[(size_t)n*21 + w*7 + v] = val;
    }
}

// ---------------------------------------------------------------------------
// Kernel 4: fold attentions into per-batch mixing matrices.
// t1v = a21·x_odd + x_even ; t2v = a12·x_even + x_odd
// n<48 : out1 = att·t1v + att2·t2v  -> U=att , Vm=att2
// n>=48: out1 = att·t2v + att2·t1v  -> U=att2, Vm=att
// Cev = U + Vm·a12_q ; Codd = Vm + U·a21_q      (3x7 each)
// out2: n<48: P=I,        Q=a21s2[q] ; n>=48: P=a12s2[q], Q=I   (3x3 each)
// ---------------------------------------------------------------------------
__global__ void combine_kernel(const float* __restrict__ att,
                               const float* __restrict__ att2,
                               const float* __restrict__ a12s1,
                               const float* __restrict__ a21s1,
                               const float* __restrict__ a12s2,
                               const float* __restrict__ a21s2,
                               float* __restrict__ Cev, float* __restrict__ Codd,
                               float* __restrict__ P,   float* __restrict__ Q)
{
    const int n = blockIdx.x;
    const bool first = (n < 48);
    const int q = first ? n : n - 48;
    const float* U  = (first ? att  : att2) + (size_t)n * 21;
    const float* Vm = (first ? att2 : att ) + (size_t)n * 21;
    const int tid = threadIdx.x;
    if (tid < 21) {
        int w = tid / 7, u = tid % 7;
        float ce = U[w*7 + u];
        float co = Vm[w*7 + u];
        for (int v = 0; v < 7; ++v) {
            ce += Vm[w*7 + v] * a12s1[(size_t)q*49 + v*7 + u];
            co += U [w*7 + v] * a21s1[(size_t)q*49 + v*7 + u];
        }
        Cev [(size_t)n*21 + tid] = ce;
        Codd[(size_t)n*21 + tid] = co;
    }
    if (tid < 9) {
        int w = tid / 3, u = tid % 3;
        float id = (w == u) ? 1.f : 0.f;
        P[(size_t)n*9 + tid] = first ? id : a12s2[(size_t)q*9 + tid];
        Q[(size_t)n*9 + tid] = first ? a21s2[(size_t)q*9 + tid] : id;
    }
}

// ---------------------------------------------------------------------------
// Kernel 5: fused mix + spatial transpose (the d<->t swap from _to_ndtv).
// out[n,i,j,w] = sum_v Ce[n,w,v]*x[2q,j,i,v] + Co[n,w,v]*x[2q+1,j,i,v]
// One block: 32x32 spatial tile, BOTH n-halves (q and q+48) which share reads.
// LDS-staged so reads (stride NVIN along i) and writes (96 contiguous floats
// per out row) are both coalesced.
// ---------------------------------------------------------------------------
template<int NVIN>
__global__ __launch_bounds__(256)
void out_mix_kernel(const float* __restrict__ x,    // (96,256,256,NVIN)
                    const float* __restrict__ Ce,   // (96,3,NVIN)
                    const float* __restrict__ Co,   // (96,3,NVIN)
                    float* __restrict__ out)        // (96,256,256,3)
{
    const int q  = blockIdx.z;
    const int i0 = blockIdx.x * 32;
    const int j0 = blockIdx.y * 32;
    __shared__ float so[2][32][97];                 // pad: conflict-free
    __shared__ float mat[2][2][3*NVIN];             // [half][even/odd]
    if (threadIdx.x < 2*2*3*NVIN) {
        int h  = threadIdx.x / (2*3*NVIN);
        int r  = threadIdx.x % (2*3*NVIN);
        int eo = r / (3*NVIN);
        int e  = r % (3*NVIN);
        int n  = q + h*48;
        mat[h][eo][e] = (eo ? Co : Ce)[(size_t)n*3*NVIN + e];
    }
    __syncthreads();

    const int il  = threadIdx.x & 31;   // i (t of input, dd of output)
    const int jls = threadIdx.x >> 5;
#pragma unroll
    for (int pass = 0; pass < 4; ++pass) {
        int jl = jls + pass*8;
        int j = j0 + jl;                // d of input, tt of output
        int i = i0 + il;
        const float* pe = x + (((size_t)(2*q)  *256 + j)*256 + i)*NVIN;
        const float* po = x + (((size_t)(2*q+1)*256 + j)*256 + i)*NVIN;
        float fe[NVIN], fo[NVIN];
#pragma unroll
        for (int v = 0; v < NVIN; ++v) { fe[v] = pe[v]; fo[v] = po[v]; }
#pragma unroll
        for (int h = 0; h < 2; ++h) {
#pragma unroll
            for (int w = 0; w < 3; ++w) {
                float s = 0.f;
#pragma unroll
                for (int v = 0; v < NVIN; ++v)
                    s += mat[h][0][w*NVIN+v]*fe[v] + mat[h][1][w*NVIN+v]*fo[v];
                so[h][il][jl*3 + w] = s;
            }
        }
    }
    __syncthreads();

#pragma unroll
    for (int h = 0; h < 2; ++h) {
        int n = q + h*48;
        float* ob = out + (((size_t)n*256 + i0)*256 + j0)*3;
        for (int e = threadIdx.x; e < 32*96; e += 256) {
            int ir = e / 96, c = e % 96;
            ob[(size_t)ir*768 + c] = so[h][ir][c];
        }
    }
}

// ---------------------------------------------------------------------------
extern "C" void kernel_launch(void* const* d_in, const int* in_sizes, int n_in,
                              void* d_out, int out_size, void* d_ws, size_t ws_size,
                              hipStream_t stream)
{
    (void)in_sizes; (void)n_in; (void)out_size; (void)ws_size;
    const float* x_s1    = (const float*)d_in[0];
    const float* x_s2    = (const float*)d_in[1];
    const float* A_j2p   = (const float*)d_in[2];
    const float* A_s1    = (const float*)d_in[3];
    const float* A_s2    = (const float*)d_in[4];
    const float* weight  = (const float*)d_in[5];
    const float* weight2 = (const float*)d_in[6];
    const float* w12s1   = (const float*)d_in[7];
    const float* w21s1   = (const float*)d_in[8];
    const float* w12s2   = (const float*)d_in[9];
    const float* w21s2   = (const float*)d_in[10];
    const float* We1     = (const float*)d_in[11];
    const float* We2     = (const float*)d_in[12];

    float* ws    = (float*)d_ws;               // ~4.5 MB total scratch
    float* part1 = ws;                         // 8 * 672*128 = 688128
    float* part2 = part1 + (size_t)KSPLIT*672*128;   // 8 * 288*128 = 294912
    float* xa1   = part2 + (size_t)KSPLIT*288*128;   // 96*7*128 = 86016
    float* xa2   = xa1   + 86016;              // 96*3*128 = 36864
    float* a12s1 = xa2   + 36864;              // 48*49
    float* a21s1 = a12s1 + 2352;
    float* a12s2 = a21s1 + 2352;               // 48*9
    float* a21s2 = a12s2 + 432;
    float* attJ  = a21s2 + 432;                // 96*21
    float* att2J = attJ  + 2016;
    float* Cev   = att2J + 2016;               // 96*21
    float* Codd  = Cev   + 2016;
    float* Pm    = Codd  + 2016;               // 96*9
    float* Qm    = Pm    + 864;

    float* out1 = (float*)d_out;
    float* out2 = out1 + (size_t)96*256*256*3;

    xa_gemm_kernel<7><<<dim3(42, KSPLIT), 256, 0, stream>>>(x_s1, We1, part1);
    xa_gemm_kernel<3><<<dim3(18, KSPLIT), 256, 0, stream>>>(x_s2, We2, part2);
    xa_reduce_kernel<<<(672*128 + 255)/256, 256, 0, stream>>>(part1, xa1, 672*128);
    xa_reduce_kernel<<<(288*128 + 255)/256, 256, 0, stream>>>(part2, xa2, 288*128);
    pair_att_kernel<7><<<48, 128, 0, stream>>>(xa1, w12s1, w21s1, A_s1, a12s1, a21s1);
    pair_att_kernel<3><<<48, 128, 0, stream>>>(xa2, w12s2, w21s2, A_s2, a12s2, a21s2);
    joint_att_kernel<<<96, 128, 0, stream>>>(xa1, xa2, weight, weight2, A_j2p, attJ, att2J);
    combine_kernel<<<96, 32, 0, stream>>>(attJ, att2J, a12s1, a21s1, a12s2, a21s2,
                                          Cev, Codd, Pm, Qm);
    out_mix_kernel<7><<<dim3(8,8,48), 256, 0, stream>>>(x_s1, Cev, Codd, out1);
    out_mix_kernel<3><<<dim3(8,8,48), 256, 0, stream>>>(x_s2, Pm, Qm, out2);
}